// OTAttention_15650860827508
// MI455X (gfx1250) — compile-verified
//
#include <hip/hip_runtime.h>
#include <math.h>

typedef __attribute__((ext_vector_type(2))) float v2f;
typedef __attribute__((ext_vector_type(4))) float v4f;
typedef __attribute__((ext_vector_type(8))) float v8f;

// Native gfx1250 base-2 transcendentals (v_exp_f32 / v_log_f32)
#define EXP2F(x) __builtin_amdgcn_exp2f(x)
#define LOG2F(x) __builtin_amdgcn_logf(x)

#define BB 8
#define NN 1024
#define DD 64
#define EPSI 0.05f
#define THRESH 1e-6f
#define MAX_ITER 100
// Everything in the Sinkhorn loop lives in the log2 domain (v_exp_f32 /
// v_log_f32 are natively base-2 -> saves one VALU mul per transcendental).
// St = (Q.K^T - 1)/eps * log2(e);  20 * log2(e):
#define INV_EPS_L2E 28.8539008f
// log2(1/1024 + 1e-8)
#define LOG2_MU -9.9999852f
// err scale back to natural-log u units: eps * ln2
#define ERR_SCALE 0.0346573590f

// ---------------------------------------------------------------------------
// 1) L2-normalize q and k rows (dim=-1), matching F.normalize semantics.
// ---------------------------------------------------------------------------
__global__ void ot_normalize_kernel(const float* __restrict__ q,
                                    const float* __restrict__ k,
                                    float* __restrict__ Qn,
                                    float* __restrict__ Kn) {
  __shared__ float sm[256];
  int tid = threadIdx.x;
  int rloc = tid >> 6;
  int d = tid & 63;
  int row = blockIdx.x * 4 + rloc;   // 0 .. 2*B*N-1
  const float* src;
  float* dst;
  int r2;
  if (row < BB * NN) { src = q; dst = Qn; r2 = row; }
  else               { src = k; dst = Kn; r2 = row - BB * NN; }
  float x = src[(size_t)r2 * DD + d];
  sm[tid] = x * x;
  __syncthreads();
  for (int off = 32; off >= 1; off >>= 1) {
    if (d < off) sm[tid] += sm[tid + off];
    __syncthreads();
  }
  float n = fmaxf(sqrtf(sm[rloc << 6]), 1e-12f);
  dst[(size_t)r2 * DD + d] = x / n;
}

// ---------------------------------------------------------------------------
// 2) St = ((Qn*Kn^T) - 1) * (log2e/eps) via V_WMMA_F32_16X16X4_F32.
//    One wave per 16x16 tile; A and B fragments are contiguous float2 loads
//    from row-major Qn / Kn (B = Kn^T).
// ---------------------------------------------------------------------------
__global__ void ot_gemm_s_kernel(const float* __restrict__ Qn,
                                 const float* __restrict__ Kn,
                                 float* __restrict__ S) {
  int lane = threadIdx.x & 31;
  int wave = threadIdx.x >> 5;
  int tile = blockIdx.x * 8 + wave;   // 0 .. B*64*64-1
  int b = tile >> 12;
  int rem = tile & 4095;
  int i0 = (rem >> 6) << 4;
  int j0 = (rem & 63) << 4;
  int half = lane >> 4, lm = lane & 15;
  const float* Qb = Qn + (size_t)b * NN * DD;
  const float* Kb = Kn + (size_t)b * NN * DD;
  v8f c = {};
  #pragma unroll
  for (int kk = 0; kk < DD; kk += 4) {
    int kh = kk + 2 * half;
    v2f a  = *(const v2f*)(Qb + (size_t)(i0 + lm) * DD + kh);
    v2f bm = *(const v2f*)(Kb + (size_t)(j0 + lm) * DD + kh);
    c = __builtin_amdgcn_wmma_f32_16x16x4_f32(false, a, false, bm,
                                              (short)0, c, false, false);
  }
  float* Sb = S + (size_t)b * NN * NN;
  #pragma unroll
  for (int r = 0; r < 8; ++r) {
    int i = i0 + r + 8 * half;
    Sb[(size_t)i * NN + (j0 + lm)] = c[r] * INV_EPS_L2E - INV_EPS_L2E;
  }
}

// ---------------------------------------------------------------------------
// 3) init potentials + flags
// ---------------------------------------------------------------------------
__global__ void ot_init_kernel(float* ut, float* vt, float* err, int* done) {
  int t = blockIdx.x * blockDim.x + threadIdx.x;
  if (t < BB * NN) { ut[t] = 0.0f; vt[t] = 0.0f; }
  if (t == 0) { *err = 0.0f; *done = 0; }
}

// ---------------------------------------------------------------------------
// 4) Row pass: ut[b,i] = LOG2_MU - log2sumexp2_j(St[b,i,j] + vt[b,j])
//    b128 loads (4 cols/thread) + wave32 shuffle reductions; 2 barriers total.
// ---------------------------------------------------------------------------
__global__ void ot_row_lse_kernel(const float* __restrict__ S,
                                  const float* __restrict__ vt,
                                  float* __restrict__ ut,
                                  float* __restrict__ err,
                                  const int* __restrict__ done) {
  if (*done) return;
  __shared__ float smM[8], smS[8];
  int row = blockIdx.x;            // 0 .. B*N-1
  int b = row >> 10;
  int tid = threadIdx.x;
  int lane = tid & 31, wid = tid >> 5;
  const float* Sr = S + (size_t)row * NN;
  const float* vb = vt + b * NN;
  v4f sv = *(const v4f*)(Sr + 4 * tid);
  v4f vv = *(const v4f*)(vb + 4 * tid);
  float x0 = sv.x + vv.x, x1 = sv.y + vv.y;
  float x2 = sv.z + vv.z, x3 = sv.w + vv.w;
  float m = fmaxf(fmaxf(x0, x1), fmaxf(x2, x3));
  #pragma unroll
  for (int off = 16; off >= 1; off >>= 1)
    m = fmaxf(m, __shfl_xor(m, off, 32));
  if (lane == 0) smM[wid] = m;
  __syncthreads();
  float M = smM[0];
  #pragma unroll
  for (int w = 1; w < 8; ++w) M = fmaxf(M, smM[w]);
  float s = EXP2F(x0 - M) + EXP2F(x1 - M) + EXP2F(x2 - M) + EXP2F(x3 - M);
  #pragma unroll
  for (int off = 16; off >= 1; off >>= 1)
    s += __shfl_xor(s, off, 32);
  if (lane == 0) smS[wid] = s;
  __syncthreads();
  if (tid == 0) {
    float tot = smS[0] + smS[1] + smS[2] + smS[3]
              + smS[4] + smS[5] + smS[6] + smS[7];
    float un = LOG2_MU - (M + LOG2F(tot));
    atomicAdd(err, fabsf(un - ut[row]) * ERR_SCALE);
    ut[row] = un;
  }
}

// ---------------------------------------------------------------------------
// 5) Col pass: vt[b,j] = LOG2_MU - log2sumexp2_i(St[b,i,j] + ut[b,i])
//    512-thread blocks: 16 row-groups x 32 lanes, 4 consecutive cols/lane
//    (b128 loads). Online log2sumexp2 state merged once through LDS.
// ---------------------------------------------------------------------------
__global__ void ot_col_lse_kernel(const float* __restrict__ S,
                                  const float* __restrict__ ut,
                                  float* __restrict__ vt,
                                  const int* __restrict__ done) {
  if (*done) return;
  __shared__ v4f smm[512], sms[512];
  int tid = threadIdx.x;           // 0..511
  int g = tid >> 5;                // row group 0..15
  int l = tid & 31;
  int b = blockIdx.x >> 3;
  int j = ((blockIdx.x & 7) << 7) + 4 * l;   // 128-col strip per block
  const float* Sb = S + (size_t)b * NN * NN;
  const float* ub = ut + b * NN;
  v4f m = {-3.4e38f, -3.4e38f, -3.4e38f, -3.4e38f};
  v4f s = {0.0f, 0.0f, 0.0f, 0.0f};
  for (int i = g; i < NN; i += 16) {
    v4f sv = *(const v4f*)(Sb + (size_t)i * NN + j);
    float ui = ub[i];
    #pragma unroll
    for (int c = 0; c < 4; ++c) {
      float x = sv[c] + ui;
      if (x > m[c]) { s[c] = s[c] * EXP2F(m[c] - x) + 1.0f; m[c] = x; }
      else          { s[c] += EXP2F(x - m[c]); }
    }
  }
  smm[tid] = m; sms[tid] = s;
  __syncthreads();
  if (g == 0) {
    #pragma unroll
    for (int rr = 1; rr < 16; ++rr) {
      v4f m2 = smm[l + 32 * rr], s2 = sms[l + 32 * rr];
      #pragma unroll
      for (int c = 0; c < 4; ++c) {
        float M = fmaxf(m[c], m2[c]);
        s[c] = s[c] * EXP2F(m[c] - M) + s2[c] * EXP2F(m2[c] - M);
        m[c] = M;
      }
    }
    v4f r;
    #pragma unroll
    for (int c = 0; c < 4; ++c) r[c] = LOG2_MU - (m[c] + LOG2F(s[c]));
    *(v4f*)(vt + b * NN + j) = r;
  }
}

// ---------------------------------------------------------------------------
// 6) convergence check (replicates torch early-break freeze)
// ---------------------------------------------------------------------------
__global__ void ot_check_kernel(float* err, int* done) {
  if (*done == 0) {
    if (*err / (float)(BB * NN) < THRESH) *done = 1;
  }
  *err = 0.0f;
}

// ---------------------------------------------------------------------------
// 7) out = exp2(St + u2 + v2) @ V + V  via WMMA f32 16x16x4.
//    One wave per (b, 16-row tile); 4 accumulators cover D=64; the A-frag
//    (transport plan) is produced on the fly with v_exp_f32.
// ---------------------------------------------------------------------------
__global__ void ot_out_kernel(const float* __restrict__ S,
                              const float* __restrict__ ut,
                              const float* __restrict__ vt,
                              const float* __restrict__ Vv,
                              float* __restrict__ out) {
  int lane = threadIdx.x & 31;
  int wave = threadIdx.x >> 5;
  int wid = blockIdx.x * 8 + wave;   // 0 .. B*64-1
  int b = wid >> 6;
  int i0 = (wid & 63) << 4;
  int half = lane >> 4, lm = lane & 15;
  const float* Sb = S + (size_t)b * NN * NN;
  const float* ub = ut + b * NN;
  const float* vb = vt + b * NN;
  const float* Vb = Vv + (size_t)b * NN * DD;
  float ui = ub[i0 + lm];
  v8f c0 = {}, c1 = {}, c2 = {}, c3 = {};
  for (int kk = 0; kk < NN; kk += 4) {
    int kh = kk + 2 * half;
    v2f sv = *(const v2f*)(Sb + (size_t)(i0 + lm) * NN + kh);
    v2f a;
    a.x = EXP2F(sv.x + ui + vb[kh]);
    a.y = EXP2F(sv.y + ui + vb[kh + 1]);
    const float* Vr0 = Vb + (size_t)kh * DD;
    const float* Vr1 = Vr0 + DD;
    v2f b0, b1, b2, b3;
    b0.x = Vr0[lm];      b0.y = Vr1[lm];
    b1.x = Vr0[lm + 16]; b1.y = Vr1[lm + 16];
    b2.x = Vr0[lm + 32]; b2.y = Vr1[lm + 32];
    b3.x = Vr0[lm + 48]; b3.y = Vr1[lm + 48];
    c0 = __builtin_amdgcn_wmma_f32_16x16x4_f32(false, a, false, b0, (short)0, c0, false, false);
    c1 = __builtin_amdgcn_wmma_f32_16x16x4_f32(false, a, false, b1, (short)0, c1, false, false);
    c2 = __builtin_amdgcn_wmma_f32_16x16x4_f32(false, a, false, b2, (short)0, c2, false, false);
    c3 = __builtin_amdgcn_wmma_f32_16x16x4_f32(false, a, false, b3, (short)0, c3, false, false);
  }
  #pragma unroll
  for (int r = 0; r < 8; ++r) {
    int i = i0 + r + 8 * half;
    float* orow = out + ((size_t)b * NN + i) * DD;
    const float* vrow = Vb + (size_t)i * DD;
    orow[lm]      = c0[r] + vrow[lm];
    orow[lm + 16] = c1[r] + vrow[lm + 16];
    orow[lm + 32] = c2[r] + vrow[lm + 32];
    orow[lm + 48] = c3[r] + vrow[lm + 48];
  }
}

// ---------------------------------------------------------------------------
extern "C" void kernel_launch(void* const* d_in, const int* in_sizes, int n_in,
                              void* d_out, int out_size, void* d_ws, size_t ws_size,
                              hipStream_t stream) {
  const float* q = (const float*)d_in[0];
  const float* k = (const float*)d_in[1];
  const float* V = (const float*)d_in[2];
  float* out = (float*)d_out;

  float* ws = (float*)d_ws;
  float* Qn = ws;                                   // B*N*D
  float* Kn = Qn + (size_t)BB * NN * DD;            // B*N*D
  float* S  = Kn + (size_t)BB * NN * DD;            // B*N*N (32 MB, L2-resident)
  float* ut = S + (size_t)BB * NN * NN;             // B*N  (log2-domain u/eps)
  float* vt = ut + BB * NN;                         // B*N  (log2-domain v/eps)
  float* err = vt + BB * NN;                        // 1
  int* done = (int*)(err + 1);                      // 1

  ot_normalize_kernel<<<2 * BB * NN / 4, 256, 0, stream>>>(q, k, Qn, Kn);
  ot_gemm_s_kernel<<<BB * 64 * 64 / 8, 256, 0, stream>>>(Qn, Kn, S);
  ot_init_kernel<<<(BB * NN + 255) / 256, 256, 0, stream>>>(ut, vt, err, done);

  for (int it = 0; it < MAX_ITER; ++it) {
    ot_row_lse_kernel<<<BB * NN, 256, 0, stream>>>(S, vt, ut, err, done);
    ot_col_lse_kernel<<<BB * 8, 512, 0, stream>>>(S, ut, vt, done);
    ot_check_kernel<<<1, 1, 0, stream>>>(err, done);
  }

  ot_out_kernel<<<BB * 64 / 8, 256, 0, stream>>>(S, ut, vt, V, out);
}